// HashEmbedder2D_16071767622190
// MI455X (gfx1250) — compile-verified
//
#include <hip/hip_runtime.h>
#include <stdint.h>

// ---------------- compile-time level constants (match numpy f64 math) -------
#define NLEVELS 16
#define HSIZE   524288u          // 2^19 entries per level
#define P1U     2654435761u
#define P2U     805459861u

constexpr int      kRes[NLEVELS]   = {16,22,30,42,58,80,111,153,212,294,406,561,776,1072,1482,2047};
constexpr uint32_t kTsize[NLEVELS] = {256u,484u,900u,1764u,3364u,6400u,12321u,23409u,44944u,86436u,
                                      164836u,314721u,524288u,524288u,524288u,524288u};
// levels 0..4 staged in LDS: 2*(256+484+900+1764+3364) = 13536 floats = 54144 B
constexpr int kStaged = 5;
constexpr int kSBase[NLEVELS]  = {0,512,1480,3280,6808, 0,0,0,0,0,0,0,0,0,0,0};   // float offsets
constexpr int kSCount[NLEVELS] = {512,968,1800,3528,6728, 0,0,0,0,0,0,0,0,0,0,0}; // floats
constexpr int kSTotal = 13536;

typedef uint32_t v4u __attribute__((ext_vector_type(4)));
typedef int      v4i __attribute__((ext_vector_type(4)));
typedef int      v8i __attribute__((ext_vector_type(8)));
typedef float    v2f __attribute__((ext_vector_type(2)));
typedef float    v4f __attribute__((ext_vector_type(4)));

__device__ __forceinline__ float clamp01(float v) { return fminf(fmaxf(v, 0.0f), 1.0f); }

__global__ __launch_bounds__(256) void
hashenc2d_kernel(const float* __restrict__ x, const float* __restrict__ tables,
                 float* __restrict__ out, int n)
{
    __shared__ float s_tab[kSTotal];

    // ---- stage coarse-level tables into LDS via Tensor Data Mover ---------
#if __has_builtin(__builtin_amdgcn_tensor_load_to_lds) && __has_builtin(__builtin_amdgcn_s_wait_tensorcnt)
    if (threadIdx.x == 0) {
        #pragma unroll
        for (int l = 0; l < kStaged; ++l) {
            uint64_t ga   = (uint64_t)(uintptr_t)(tables + (size_t)l * (2u * HSIZE));
            uint32_t la   = (uint32_t)(uintptr_t)(void*)&s_tab[kSBase[l]];
            uint32_t nelm = (uint32_t)kSCount[l];        // 4-byte elements, < 65536
            // D# group 0: count=1, lds_addr, global_addr[56:0], type=2
            v4u g0 = { 1u, la, (uint32_t)ga,
                       (uint32_t)(((ga >> 32) & 0x01FFFFFFu) | (2u << 30)) };
            // D# group 1: data_size=2 (4B); 1-D tile: tensor_dim0=tile_dim0=nelm
            v8i g1 = { (int)(2u << 16),
                       (int)((nelm & 0xFFFFu) << 16),    // tensor_dim0[15:0]  @bits 63:48
                       (int)((nelm >> 16) | (1u << 16)), // tensor_dim0[31:16], tensor_dim1=1
                       (int)(nelm << 16),                // tile_dim0          @bits 127:112
                       0,                                // tile_dim1=0, tile_dim2=0
                       (int)nelm,                        // tensor_dim0_stride[31:0]
                       0, 0 };
            v4i z4 = {0,0,0,0};
#if defined(__clang_major__) && (__clang_major__ >= 23)
            v8i z8 = {0,0,0,0,0,0,0,0};
            __builtin_amdgcn_tensor_load_to_lds(g0, g1, z4, z4, z8, 0);
#else
            __builtin_amdgcn_tensor_load_to_lds(g0, g1, z4, z4, 0);
#endif
        }
        __builtin_amdgcn_s_wait_tensorcnt(0);
    }
    __syncthreads();
#else
    // fallback: cooperative vector copy (all counts divisible by 4)
    #pragma unroll
    for (int l = 0; l < kStaged; ++l) {
        const float4* src = (const float4*)(tables + (size_t)l * (2u * HSIZE));
        float4*       dst = (float4*)&s_tab[kSBase[l]];
        for (int j = threadIdx.x; j < kSCount[l] / 4; j += blockDim.x) dst[j] = src[j];
    }
    __syncthreads();
#endif

    // ---- main grid-stride loop: one thread per point ----------------------
    const int stride = gridDim.x * blockDim.x;
    for (int i = blockIdx.x * blockDim.x + threadIdx.x; i < n; i += stride) {
        __builtin_prefetch(x + 2 * (size_t)(i + stride), 0, 0);  // global_prefetch

        v2f xin = __builtin_nontemporal_load((const v2f*)x + i);  // streamed once: NT
        float xn = clamp01((xin.x + 1.0f) * 0.5f);
        float yn = clamp01((xin.y + 1.0f) * 0.5f);

        float acc[2 * NLEVELS];

        #pragma unroll
        for (int l = 0; l < NLEVELS; ++l) {
            const float rf = (float)kRes[l];
            const int rmax = kRes[l] - 1;
            float xs = xn * rf, ys = yn * rf;
            float xf = floorf(xs), yf = floorf(ys);
            float wx = xs - xf,  wy = ys - yf;     // weights from UNclamped floor (matches ref)
            int x0 = min((int)xf, rmax);
            int y0 = min((int)yf, rmax);
            int x1 = min(x0 + 1, rmax);
            int y1 = min(y0 + 1, rmax);

            // Exact int64 hash via 32-bit halves:  h = hi*2^32 + lo,  hi < 2^11.
            // h mod t == (hi*(2^32 mod t) + lo mod t) mod t, sum < 2^30+2^19 (no overflow).
            const uint32_t t   = kTsize[l];                                  // unroll-constant
            const uint32_t r32 = (uint32_t)(0x100000000ull % (uint64_t)kTsize[l]); // folds; 0 for pow2
            uint32_t axl0 = (uint32_t)x0 * P1U, axh0 = __umulhi((uint32_t)x0, P1U);
            uint32_t axl1 = (uint32_t)x1 * P1U, axh1 = __umulhi((uint32_t)x1, P1U);
            uint32_t ayl0 = (uint32_t)y0 * P2U, ayh0 = __umulhi((uint32_t)y0, P2U);
            uint32_t ayl1 = (uint32_t)y1 * P2U, ayh1 = __umulhi((uint32_t)y1, P2U);
            uint32_t c00 = ((axh0 ^ ayh0) * r32 + (axl0 ^ ayl0) % t) % t;    // const-magic urem32
            uint32_t c10 = ((axh1 ^ ayh0) * r32 + (axl1 ^ ayl0) % t) % t;
            uint32_t c01 = ((axh0 ^ ayh1) * r32 + (axl0 ^ ayl1) % t) % t;
            uint32_t c11 = ((axh1 ^ ayh1) * r32 + (axl1 ^ ayl1) % t) % t;

            v2f v00, v10, v01, v11;
            if (l < kStaged) {
                const v2f* s2 = (const v2f*)&s_tab[kSBase[l]];
                v00 = s2[c00]; v10 = s2[c10]; v01 = s2[c01]; v11 = s2[c11];
            } else {
                const v2f* t2 = (const v2f*)tables + (size_t)l * HSIZE;
                v00 = t2[c00]; v10 = t2[c10]; v01 = t2[c01]; v11 = t2[c11];
            }

            float a0 = v00.x + wx * (v10.x - v00.x);
            float a1 = v01.x + wx * (v11.x - v01.x);
            acc[2 * l + 0] = a0 + wy * (a1 - a0);
            float b0 = v00.y + wx * (v10.y - v00.y);
            float b1 = v01.y + wx * (v11.y - v01.y);
            acc[2 * l + 1] = b0 + wy * (b1 - b0);
        }

        // NT stores: keep the 192MB L2 dedicated to the resident 64MB table set
        v4f* o4 = (v4f*)(out + (size_t)i * (2 * NLEVELS));
        #pragma unroll
        for (int k = 0; k < (2 * NLEVELS) / 4; ++k) {
            v4f v = { acc[4 * k], acc[4 * k + 1], acc[4 * k + 2], acc[4 * k + 3] };
            __builtin_nontemporal_store(v, o4 + k);
        }
    }
}

extern "C" void kernel_launch(void* const* d_in, const int* in_sizes, int n_in,
                              void* d_out, int out_size, void* d_ws, size_t ws_size,
                              hipStream_t stream) {
    const float* x      = (const float*)d_in[0];   // (BATCH, 2) f32
    const float* tables = (const float*)d_in[1];   // (16, 2^19, 2) f32
    float*       out    = (float*)d_out;           // (BATCH, 32) f32
    int n = in_sizes[0] / 2;

    int blocks = (n + 255) / 256;
    if (blocks > 1024) blocks = 1024;              // grid-stride; amortizes LDS staging
    if (blocks < 1) blocks = 1;
    hipLaunchKernelGGL(hashenc2d_kernel, dim3(blocks), dim3(256), 0, stream,
                       x, tables, out, n);
}